// MultiScaleHypergraphConv_67010079752613
// MI455X (gfx1250) — compile-verified
//
#include <hip/hip_runtime.h>
#include <hip/hip_bf16.h>

typedef __attribute__((ext_vector_type(2))) float v2f;
typedef __attribute__((ext_vector_type(8))) float v8f;

static constexpr int   kNodes = 100000;
static constexpr int   kEdges = 20000;
static constexpr int   kC     = 256;
static constexpr int   kAttH  = 128;
static constexpr float kEps   = 1e-5f;

// ---------------------------------------------------------------- degrees
__global__ void degrees_kernel(const int* __restrict__ node_idx,
                               const int* __restrict__ edge_idx,
                               float* __restrict__ deg_v,
                               float* __restrict__ deg_e,
                               int nnz) {
  int i = blockIdx.x * blockDim.x + threadIdx.x;
  if (i >= nnz) return;
  atomicAdd(&deg_v[node_idx[i]], 1.0f);
  atomicAdd(&deg_e[edge_idx[i]], 1.0f);
}

__global__ void invert_kernel(float* __restrict__ d, int n) {
  int i = blockIdx.x * blockDim.x + threadIdx.x;
  if (i >= n) return;
  float v = d[i];
  d[i] = (v > 0.0f) ? (1.0f / v) : 0.0f;
}

// ----------------------------------------------- row gather / scatter-add
// 64 consecutive lanes share one nz entry and cover the 256-ch row as float4.
__global__ void scatter_rows_kernel(const float* __restrict__ src,
                                    const int* __restrict__ gather_idx,
                                    const int* __restrict__ scatter_idx,
                                    float* __restrict__ dst,
                                    int nnz) {
  long long t = (long long)blockIdx.x * blockDim.x + threadIdx.x;
  if (t >= (long long)nnz * (kC / 4)) return;
  int nz = (int)(t >> 6);
  int cg = (int)(t & 63);
  int s = gather_idx[nz];
  int d = scatter_idx[nz];
  float4 v = ((const float4*)(src + (long long)s * kC))[cg];
  float* o = dst + (long long)d * kC + cg * 4;
  atomicAdd(o + 0, v.x);
  atomicAdd(o + 1, v.y);
  atomicAdd(o + 2, v.z);
  atomicAdd(o + 3, v.w);
}

__global__ void scale_rows_kernel(float* __restrict__ data,
                                  const float* __restrict__ rs,
                                  int rows) {
  long long t = (long long)blockIdx.x * blockDim.x + threadIdx.x;
  if (t >= (long long)rows * 64) return;
  int r  = (int)(t >> 6);
  int cg = (int)(t & 63);
  float s = rs[r];
  float4* p = (float4*)(data + (long long)r * kC) + cg;
  float4 v = *p;
  v.x *= s; v.y *= s; v.z *= s; v.w *= s;
  *p = v;
}

__global__ void finalize_rows_kernel(float* __restrict__ data,
                                     const float* __restrict__ rs,
                                     const float* __restrict__ bias,
                                     int rows) {
  long long t = (long long)blockIdx.x * blockDim.x + threadIdx.x;
  if (t >= (long long)rows * 64) return;
  int r  = (int)(t >> 6);
  int cg = (int)(t & 63);
  float s = rs[r];
  float4 bv = ((const float4*)bias)[cg];
  float4* p = (float4*)(data + (long long)r * kC) + cg;
  float4 v = *p;
  v.x = v.x * s + bv.x;
  v.y = v.y * s + bv.y;
  v.z = v.z * s + bv.z;
  v.w = v.w * s + bv.w;
  *p = v;
}

// ---------------------------------------------------------------- WMMA GEMM
// D[M,Nout] = act( A[M,K] @ B[K,Nout] + bias ), one wave per 16x16 tile,
// fp32 accumulation via V_WMMA_F32_16X16X4_F32 (K chained in steps of 4).
// AVG: A := 0.5*(A0 + A1)   (used for node_features = mean(out0,out1))
template <bool AVG, bool BIAS, bool RELU>
__global__ void wmma_gemm_kernel(const float* __restrict__ A0,
                                 const float* __restrict__ A1,
                                 const float* __restrict__ B,
                                 const float* __restrict__ bias,
                                 float* __restrict__ D,
                                 int M, int K, int Nout) {
  const int lane  = threadIdx.x;            // 0..31 (wave32)
  const int tileM = blockIdx.y * blockDim.y + threadIdx.y;
  if (tileM * 16 >= M) return;              // wave-uniform: EXEC stays all-1s
  const int rowBase = tileM * 16;
  const int colBase = blockIdx.x * 16;
  const int half = lane >> 4;               // 0: K={k,k+1}, 1: K={k+2,k+3}
  const int lid  = lane & 15;
  const int m = rowBase + lid;              // A row for this lane
  const int n = colBase + lid;              // B/D column for this lane

  const float* Ar0 = A0 + (long long)m * K;
  const float* Ar1 = AVG ? (A1 + (long long)m * K) : nullptr;

  v8f acc = {0.f, 0.f, 0.f, 0.f, 0.f, 0.f, 0.f, 0.f};
  for (int k0 = 0; k0 < K; k0 += 4) {
    const int kb = k0 + 2 * half;
    v2f a, b;
    if (AVG) {
      a.x = 0.5f * (Ar0[kb]     + Ar1[kb]);
      a.y = 0.5f * (Ar0[kb + 1] + Ar1[kb + 1]);
    } else {
      a.x = Ar0[kb];
      a.y = Ar0[kb + 1];
    }
    b.x = B[(long long)kb * Nout + n];
    b.y = B[(long long)(kb + 1) * Nout + n];
    acc = __builtin_amdgcn_wmma_f32_16x16x4_f32(
        /*neg_a=*/false, a, /*neg_b=*/false, b,
        /*c_mod=*/(short)0, acc, /*reuse_a=*/false, /*reuse_b=*/false);
  }

#pragma unroll
  for (int r = 0; r < 8; ++r) {
    const int mr = rowBase + r + 8 * half;  // C/D layout: VGPR r -> M=r / M=8+r
    float v = acc[r];
    if (BIAS) v += bias[n];
    if (RELU) v = v > 0.0f ? v : 0.0f;
    D[(long long)mr * Nout + n] = v;
  }
}

// --------------------------------------- attention layer 2 + softmax + mean
// wave per node: logits = a1[node,:] @ W2 + b2 ; softmax(2) ; block-reduced
// partial sums -> one pair of atomics per block (8 nodes).
__global__ void attn2_reduce_kernel(const float* __restrict__ a1,
                                    const float* __restrict__ W2,
                                    const float* __restrict__ b2,
                                    float* __restrict__ sums) {
  __shared__ float s0[8];
  __shared__ float s1[8];
  const int wid  = threadIdx.x >> 5;
  const int lane = threadIdx.x & 31;
  const int node = blockIdx.x * 8 + wid;
  float p0 = 0.0f, p1 = 0.0f;
  if (node < kNodes) {
    float4 av = ((const float4*)(a1 + (long long)node * kAttH))[lane];
    float4 w0 = ((const float4*)W2)[lane * 2];      // (k0c0,k0c1,k1c0,k1c1)
    float4 w1 = ((const float4*)W2)[lane * 2 + 1];  // (k2c0,k2c1,k3c0,k3c1)
    p0 = av.x * w0.x + av.y * w0.z + av.z * w1.x + av.w * w1.z;
    p1 = av.x * w0.y + av.y * w0.w + av.z * w1.y + av.w * w1.w;
  }
  for (int off = 16; off; off >>= 1) {
    p0 += __shfl_down(p0, off, 32);
    p1 += __shfl_down(p1, off, 32);
  }
  if (lane == 0) {
    if (node < kNodes) {
      float l0 = p0 + b2[0], l1 = p1 + b2[1];
      float mx = fmaxf(l0, l1);
      float e0 = __expf(l0 - mx), e1 = __expf(l1 - mx);
      float inv = 1.0f / (e0 + e1);
      s0[wid] = e0 * inv;
      s1[wid] = e1 * inv;
    } else {
      s0[wid] = 0.0f;
      s1[wid] = 0.0f;
    }
  }
  __syncthreads();
  if (threadIdx.x == 0) {
    float a = 0.0f, b = 0.0f;
#pragma unroll
    for (int i = 0; i < 8; ++i) { a += s0[i]; b += s1[i]; }
    atomicAdd(&sums[0], a);
    atomicAdd(&sums[1], b);
  }
}

__global__ void weights_kernel(const float* __restrict__ scale_w,
                               const float* __restrict__ sums,
                               float* __restrict__ w) {
  float v0 = scale_w[0], v1 = scale_w[1];
  float mx = fmaxf(v0, v1);
  float e0 = __expf(v0 - mx), e1 = __expf(v1 - mx);
  float inv = 1.0f / (e0 + e1);
  const float invN = 1.0f / (float)kNodes;
  w[0] = 0.5f * (e0 * inv + sums[0] * invN);
  w[1] = 0.5f * (e1 * inv + sums[1] * invN);
}

// ----------------------------- LayerNorm + ReLU + weighted accumulate
// wave per row (32 lanes x 8 channels), shfl_xor butterfly reduction.
template <bool ACCUM>
__global__ void ln_accum_kernel(const float* __restrict__ h,
                                const float* __restrict__ g,
                                const float* __restrict__ b,
                                const float* __restrict__ w, int wi,
                                float* __restrict__ out) {
  const int row  = blockIdx.x * 8 + (threadIdx.x >> 5);
  const int lane = threadIdx.x & 31;
  if (row >= kNodes) return;
  const float4* hr = (const float4*)(h + (long long)row * kC);

  float v[8];
  *(float4*)&v[0] = hr[lane * 2];
  *(float4*)&v[4] = hr[lane * 2 + 1];

  float s = 0.0f, ss = 0.0f;
#pragma unroll
  for (int j = 0; j < 8; ++j) { s += v[j]; ss += v[j] * v[j]; }
  for (int off = 16; off; off >>= 1) {
    s  += __shfl_xor(s, off, 32);
    ss += __shfl_xor(ss, off, 32);
  }
  const float mu   = s * (1.0f / kC);
  const float var  = ss * (1.0f / kC) - mu * mu;
  const float rstd = rsqrtf(var + kEps);
  const float ww   = w[wi];

  float gv[8], bv[8];
  *(float4*)&gv[0] = ((const float4*)g)[lane * 2];
  *(float4*)&gv[4] = ((const float4*)g)[lane * 2 + 1];
  *(float4*)&bv[0] = ((const float4*)b)[lane * 2];
  *(float4*)&bv[4] = ((const float4*)b)[lane * 2 + 1];

  float r[8];
#pragma unroll
  for (int j = 0; j < 8; ++j)
    r[j] = fmaxf((v[j] - mu) * rstd * gv[j] + bv[j], 0.0f) * ww;

  float4* orow = (float4*)(out + (long long)row * kC);
  if (ACCUM) {
    float p[8];
    *(float4*)&p[0] = orow[lane * 2];
    *(float4*)&p[4] = orow[lane * 2 + 1];
#pragma unroll
    for (int j = 0; j < 8; ++j) r[j] += p[j];
  }
  orow[lane * 2]     = *(float4*)&r[0];
  orow[lane * 2 + 1] = *(float4*)&r[4];
}

// ---------------------------------------------------------------- launch
extern "C" void kernel_launch(void* const* d_in, const int* in_sizes, int n_in,
                              void* d_out, int out_size, void* d_ws, size_t ws_size,
                              hipStream_t stream) {
  const float* x       = (const float*)d_in[0];
  const int*   hei     = (const int*)d_in[1];
  const float* conv_W  = (const float*)d_in[2];
  const float* conv_b  = (const float*)d_in[3];
  const float* tr_W    = (const float*)d_in[4];
  const float* tr_b    = (const float*)d_in[5];
  const float* ln_g    = (const float*)d_in[6];
  const float* ln_b    = (const float*)d_in[7];
  const float* scale_w = (const float*)d_in[8];
  const float* aW1     = (const float*)d_in[9];
  const float* ab1     = (const float*)d_in[10];
  const float* aW2     = (const float*)d_in[11];
  const float* ab2     = (const float*)d_in[12];
  float* out = (float*)d_out;

  const int nnz = in_sizes[1] / 2;
  const int* node_idx = hei;
  const int* edge_idx = hei + nnz;

  float* f = (float*)d_ws;
  float* deg_e  = f; f += 20480;                       // [E] -> Binv in place
  float* deg_v  = f; f += 100352;                      // [N] -> Dinv in place
  float* e_feat = f; f += (long long)kEdges * kC;      // [E,C]
  float* out0   = f; f += (long long)kNodes * kC;      // [N,C]
  float* out1   = f; f += (long long)kNodes * kC;      // [N,C]
  float* buf    = f; f += (long long)kNodes * kC;      // xt / a1 / h scratch
  float* sums   = f; f += 4;                           // attention sums
  float* wts    = f; f += 4;                           // combined weights

  // ---- degrees + inverses
  hipMemsetAsync(deg_e, 0, (size_t)kEdges * sizeof(float), stream);
  hipMemsetAsync(deg_v, 0, (size_t)kNodes * sizeof(float), stream);
  degrees_kernel<<<(nnz + 255) / 256, 256, 0, stream>>>(node_idx, edge_idx,
                                                        deg_v, deg_e, nnz);
  invert_kernel<<<(kEdges + 255) / 256, 256, 0, stream>>>(deg_e, kEdges);
  invert_kernel<<<(kNodes + 255) / 256, 256, 0, stream>>>(deg_v, kNodes);

  const dim3 gblock(32, 4);                            // 4 wave32 per block
  const int  mblocks = (kNodes / 16 + 3) / 4;          // 6250 tiles -> 1563
  const int  scat_blocks = (int)(((long long)nnz * 64 + 255) / 256);

  float* outs[2] = {out0, out1};
  for (int s = 0; s < 2; ++s) {
    // xt = x @ conv_W[s]
    wmma_gemm_kernel<false, false, false>
        <<<dim3(kC / 16, mblocks), gblock, 0, stream>>>(
            x, nullptr, conv_W + (long long)s * kC * kC, nullptr, buf,
            kNodes, kC, kC);
    // e_feat = segment_sum(xt[node_idx]) * Binv
    hipMemsetAsync(e_feat, 0, (size_t)kEdges * kC * sizeof(float), stream);
    scatter_rows_kernel<<<scat_blocks, 256, 0, stream>>>(buf, node_idx,
                                                         edge_idx, e_feat, nnz);
    scale_rows_kernel<<<(kEdges * 64 + 255) / 256, 256, 0, stream>>>(e_feat,
                                                                     deg_e,
                                                                     kEdges);
    // out = segment_sum(e_feat[edge_idx]) * Dinv + conv_b[s]
    hipMemsetAsync(outs[s], 0, (size_t)kNodes * kC * sizeof(float), stream);
    scatter_rows_kernel<<<scat_blocks, 256, 0, stream>>>(e_feat, edge_idx,
                                                         node_idx, outs[s], nnz);
    finalize_rows_kernel<<<(kNodes * 64 + 255) / 256, 256, 0, stream>>>(
        outs[s], deg_v, conv_b + (long long)s * kC, kNodes);
  }

  // ---- attention: a1 = relu(((out0+out1)/2) @ W1 + b1); dyn_w = mean softmax
  hipMemsetAsync(sums, 0, 2 * sizeof(float), stream);
  wmma_gemm_kernel<true, true, true>
      <<<dim3(kAttH / 16, mblocks), gblock, 0, stream>>>(
          out0, out1, aW1, ab1, buf, kNodes, kC, kAttH);
  attn2_reduce_kernel<<<(kNodes + 7) / 8, 256, 0, stream>>>(buf, aW2, ab2, sums);
  weights_kernel<<<1, 1, 0, stream>>>(scale_w, sums, wts);

  // ---- per-scale transform + LayerNorm + ReLU + weighted accumulate
  wmma_gemm_kernel<false, true, false>
      <<<dim3(kC / 16, mblocks), gblock, 0, stream>>>(
          out0, nullptr, tr_W, tr_b, buf, kNodes, kC, kC);
  ln_accum_kernel<false><<<(kNodes + 7) / 8, 256, 0, stream>>>(
      buf, ln_g, ln_b, wts, 0, out);

  wmma_gemm_kernel<false, true, false>
      <<<dim3(kC / 16, mblocks), gblock, 0, stream>>>(
          out1, nullptr, tr_W + (long long)kC * kC, tr_b + kC, buf,
          kNodes, kC, kC);
  ln_accum_kernel<true><<<(kNodes + 7) / 8, 256, 0, stream>>>(
      buf, ln_g + kC, ln_b + kC, wts, 1, out);
}